// GridSamplePScan_24094766530929
// MI455X (gfx1250) — compile-verified
//
#include <hip/hip_runtime.h>
#include <math.h>

typedef __attribute__((ext_vector_type(2))) float v2f;
typedef __attribute__((ext_vector_type(8))) float v8f;

#define Bn 2
#define Ln 24
#define Cn 16
#define Hn 64
#define Wn 64
#define HW   (Hn * Wn)        // 4096
#define CHW  (Cn * HW)        // 65536
#define LCHW (Ln * CHW)       // per-batch images stride
#define FROW (2 * HW)         // 8192 floats: one L-row of flows [2,H,W]
#define FB   (Ln * FROW)      // per-batch flows size

// ---------------------------------------------------------------------------
// Kernel 1: cumsum over L as lower-triangular matmul via V_WMMA_F32_16X16X4_F32
// cum = T @ F, T = 24x24 lower-triangular ones (padded to 32 rows),
// F = flows[b] viewed as [24, 8192]. Each wave computes a 32x16 output strip.
// ---------------------------------------------------------------------------
__global__ __launch_bounds__(128) void pscan_wmma_kernel(
    const float* __restrict__ flows, float* __restrict__ cum) {
  const int wave = blockIdx.x * (blockDim.x >> 5) + (threadIdx.x >> 5);
  const int lane = threadIdx.x & 31;
  const int b  = wave >> 9;       // 512 column tiles of 16 per batch
  const int jt = wave & 511;
  const int j0 = jt * 16;

  const float* F  = flows + b * FB;
  float*       Cm = cum   + b * FB;

  const int lhalf = lane >> 4;    // 0 = lanes 0..15, 1 = lanes 16..31
  const int lidx  = lane & 15;

  v8f d0 = {};   // output rows t = 0..15
  v8f d1 = {};   // output rows t = 16..31 (24..31 discarded)

  #pragma unroll
  for (int kt = 0; kt < 6; ++kt) {
    const int k0 = kt * 4;
    // B tile (4x16, K x N): VGPR0 = rows K={0,1}, VGPR1 = rows K={2,3};
    // lanes 0-15 hold the even row of the pair, lanes 16-31 the odd row.
    v2f bt;
    bt.x = F[(k0 + lhalf)     * FROW + j0 + lidx];
    bt.y = F[(k0 + 2 + lhalf) * FROW + j0 + lidx];
    // A tile (16x4, M x K) synthesized: a[m][k] = (global_k <= m) ? 1 : 0.
    // Layout: lane m (0..15 per half), VGPR0 holds K = 2*lhalf, VGPR1 K = 2*lhalf+1.
    const int ka = k0 + 2 * lhalf;
    v2f a0, a1;
    a0.x = (ka     <= lidx)      ? 1.0f : 0.0f;
    a0.y = (ka + 1 <= lidx)      ? 1.0f : 0.0f;
    a1.x = (ka     <= lidx + 16) ? 1.0f : 0.0f;
    a1.y = (ka + 1 <= lidx + 16) ? 1.0f : 0.0f;

    d0 = __builtin_amdgcn_wmma_f32_16x16x4_f32(false, a0, false, bt,
                                               (short)0, d0, false, false);
    d1 = __builtin_amdgcn_wmma_f32_16x16x4_f32(false, a1, false, bt,
                                               (short)0, d1, false, false);
  }

  // Store. 16x16 f32 D layout: VGPR r -> row r (lanes 0-15) / row r+8 (lanes 16-31).
  const int rowoff = lhalf ? 8 : 0;
  #pragma unroll
  for (int r = 0; r < 8; ++r) {
    const int t0 = r + rowoff;          // 0..15, always valid
    Cm[t0 * FROW + j0 + lidx] = d0[r];
    const int t1 = 16 + r + rowoff;     // 16..31, valid only < 24
    if (t1 < Ln) Cm[t1 * FROW + j0 + lidx] = d1[r];
  }
}

// ---------------------------------------------------------------------------
// Kernel 2: per-(b,t,pixel) thread loops k = 0..t, computes warp coords once
// per k, bilinear-gathers all 16 channels into register accumulators.
// Segment-sum over k happens in registers; no atomics needed.
// ---------------------------------------------------------------------------
__global__ __launch_bounds__(256) void warp_accum_kernel(
    const float* __restrict__ images, const float* __restrict__ cum,
    float* __restrict__ out) {
  const int blk   = blockIdx.x;         // B * L * 16 blocks
  const int chunk = blk & 15;
  const int bt    = blk >> 4;
  const int t = bt % Ln;
  const int b = bt / Ln;
  const int p = chunk * 256 + threadIdx.x;   // pixel 0..4095
  const int h = p >> 6;
  const int w = p & 63;

  const float* cumb = cum + b * FB;
  const float  ctx  = cumb[t * FROW + p];        // cum[t], x component
  const float  cty  = cumb[t * FROW + HW + p];   // cum[t], y component

  const float gx = (2.0f * (float)w + 1.0f) * (1.0f / (float)Wn) - 1.0f;
  const float gy = (2.0f * (float)h + 1.0f) * (1.0f / (float)Hn) - 1.0f;

  float acc[Cn];
  #pragma unroll
  for (int c = 0; c < Cn; ++c) acc[c] = 0.0f;

  const float* imb = images + b * LCHW;

  for (int k = 0; k <= t; ++k) {
    const float relx = ctx - cumb[k * FROW + p];
    const float rely = cty - cumb[k * FROW + HW + p];

    // wrap x into [-1, 1): remainder with positive divisor semantics
    float r = fmodf(gx + relx + 1.0f, 2.0f);
    r = (r < 0.0f) ? r + 2.0f : r;
    const float gxw = r - 1.0f;

    const float ix = (gxw + 1.0f)       * ((float)Wn * 0.5f) - 0.5f;
    const float iy = (gy + rely + 1.0f) * ((float)Hn * 0.5f) - 0.5f;

    const float x0f = floorf(ix), y0f = floorf(iy);
    const float x1f = x0f + 1.0f, y1f = y0f + 1.0f;
    const float wx1 = ix - x0f, wy1 = iy - y0f;
    const float wx0 = 1.0f - wx1, wy0 = 1.0f - wy1;

    const float vx0 = (x0f >= 0.0f && x0f <= (float)(Wn - 1)) ? 1.0f : 0.0f;
    const float vx1 = (x1f >= 0.0f && x1f <= (float)(Wn - 1)) ? 1.0f : 0.0f;
    const float vy0 = (y0f >= 0.0f && y0f <= (float)(Hn - 1)) ? 1.0f : 0.0f;
    const float vy1 = (y1f >= 0.0f && y1f <= (float)(Hn - 1)) ? 1.0f : 0.0f;

    int xi0 = (int)x0f; xi0 = xi0 < 0 ? 0 : (xi0 > Wn - 1 ? Wn - 1 : xi0);
    int xi1 = (int)x1f; xi1 = xi1 < 0 ? 0 : (xi1 > Wn - 1 ? Wn - 1 : xi1);
    int yi0 = (int)y0f; yi0 = yi0 < 0 ? 0 : (yi0 > Hn - 1 ? Hn - 1 : yi0);
    int yi1 = (int)y1f; yi1 = yi1 < 0 ? 0 : (yi1 > Hn - 1 ? Hn - 1 : yi1);

    const float w00 = wx0 * wy0 * vx0 * vy0;
    const float w01 = wx1 * wy0 * vx1 * vy0;
    const float w10 = wx0 * wy1 * vx0 * vy1;
    const float w11 = wx1 * wy1 * vx1 * vy1;

    const int o00 = yi0 * Wn + xi0;
    const int o01 = yi0 * Wn + xi1;
    const int o10 = yi1 * Wn + xi0;
    const int o11 = yi1 * Wn + xi1;

    const float* ik = imb + k * CHW;
    #pragma unroll
    for (int c = 0; c < Cn; ++c) {
      const float* ic = ik + c * HW;
      float a = acc[c];
      a = fmaf(w00, ic[o00], a);
      a = fmaf(w01, ic[o01], a);
      a = fmaf(w10, ic[o10], a);
      a = fmaf(w11, ic[o11], a);
      acc[c] = a;
    }
  }

  float* ob = out + b * LCHW + t * CHW;
  #pragma unroll
  for (int c = 0; c < Cn; ++c) ob[c * HW + p] = acc[c];
}

extern "C" void kernel_launch(void* const* d_in, const int* in_sizes, int n_in,
                              void* d_out, int out_size, void* d_ws, size_t ws_size,
                              hipStream_t stream) {
  const float* flows  = (const float*)d_in[0];   // [B, L, 2, H, W] f32
  const float* images = (const float*)d_in[1];   // [B, L, C, H, W] f32
  float*       out    = (float*)d_out;           // [B, L, C, H, W] f32
  float*       cum    = (float*)d_ws;            // B*L*2*H*W floats = 1.5 MB

  // Stage 1: prefix scan over L via triangular-matrix WMMA.
  // 1024 waves total (2 batches x 512 column-tiles), 4 waves per block.
  pscan_wmma_kernel<<<256, 128, 0, stream>>>(flows, cum);

  // Stage 2: warp + in-register segment-sum. B*L*16 blocks x 256 threads.
  warp_accum_kernel<<<Bn * Ln * 16, 256, 0, stream>>>(images, cum, out);
}